// HopfieldNetwork_50328426774856
// MI455X (gfx1250) — compile-verified
//
#include <hip/hip_runtime.h>
#include <hip/hip_bf16.h>
#include <stdint.h>

#define N_NEURONS 6400
#define BATCH 8

typedef float v2f __attribute__((ext_vector_type(2)));
typedef float v8f __attribute__((ext_vector_type(8)));
typedef unsigned int u32x4 __attribute__((ext_vector_type(4)));
typedef int i32x4 __attribute__((ext_vector_type(4)));
typedef int i32x8 __attribute__((ext_vector_type(8)));

// ---- order-preserving float<->uint key for atomic max over signed floats ----
__device__ __forceinline__ unsigned int float_key(float f) {
    unsigned int b = __float_as_uint(f);
    return (b & 0x80000000u) ? ~b : (b | 0x80000000u);
}
__device__ __forceinline__ float key_float(unsigned int k) {
    unsigned int b = (k & 0x80000000u) ? (k ^ 0x80000000u) : ~k;
    return __uint_as_float(b);
}

__device__ __forceinline__ unsigned int wang_hash(unsigned int s) {
    s = (s ^ 61u) ^ (s >> 16);
    s *= 9u;
    s = s ^ (s >> 4);
    s *= 0x27d4eb2du;
    s = s ^ (s >> 15);
    return s;
}

// ---------------------------------------------------------------------------
// TDM: 1-D contiguous tile load Global -> LDS (CDNA5 Tensor Data Mover).
// Descriptor bitfields per CDNA5 ISA (group0) / §8.4
// (group1). count=1, type=2("image"), data_size=2 (4 bytes),
// tensor_dim0 = tile_dim0 = nelems, tensor_dim1 = tile_dim1 = 1.
// This toolchain exposes the 6-arg builtin:
//   (u32x4 g0, i32x8 g1, i32x4 g2, i32x4 g3, i32x8 extra, i32 cpol)
// ---------------------------------------------------------------------------
__device__ __forceinline__ void tdm_load_1d(unsigned int lds_addr,
                                            const void* gptr,
                                            unsigned int nelems) {
#if __has_builtin(__builtin_amdgcn_tensor_load_to_lds)
    const unsigned long long ga = (unsigned long long)gptr;
    u32x4 g0;
    g0[0] = 1u;                                     // count=1, is_restore=0
    g0[1] = lds_addr;                               // lds_addr[31:0]
    g0[2] = (unsigned int)ga;                       // global_addr[31:0] (bits 95:64)
    g0[3] = ((unsigned int)(ga >> 32) & 0x01FFFFFFu)// global_addr[56:32]
          | 0x80000000u;                            // type=2 (bits 127:126)
    i32x8 g1;
    g1[0] = (int)(2u << 16);                        // data_size=2 -> 4 bytes
    g1[1] = (int)((nelems & 0xFFFFu) << 16);        // tensor_dim0[15:0]
    g1[2] = (int)(((nelems >> 16) & 0xFFFFu)        // tensor_dim0[31:16]
          | (1u << 16));                            // tensor_dim1 = 1
    g1[3] = (int)((nelems & 0xFFFFu) << 16);        // tile_dim0 (tensor_dim1 hi = 0)
    g1[4] = 1;                                      // tile_dim1=1, tile_dim2=0
    g1[5] = (int)nelems;                            // tensor_dim0_stride[31:0]
    g1[6] = 0;                                      // stride hi / dim1_stride lo
    g1[7] = 0;
    i32x4 gz4 = {0, 0, 0, 0};
    i32x8 gz8 = {0, 0, 0, 0, 0, 0, 0, 0};
    __builtin_amdgcn_tensor_load_to_lds(g0, g1, gz4, gz4, gz8, 0 /*cpol*/);
#else
    (void)lds_addr; (void)gptr; (void)nelems;
#endif
}

// ---------------------------------------------------------------------------
// Kernel 0: init workspace (maxkey = 0, the smallest key)
// ---------------------------------------------------------------------------
__global__ void init_ws_kernel(unsigned int* maxkey) {
    if (threadIdx.x == 0) *maxkey = 0u;
}

// ---------------------------------------------------------------------------
// Kernel 1: global max over W (164 MB single pass; also warms the 192MB L2,
// which then holds essentially all of W for the rest of the pipeline)
// ---------------------------------------------------------------------------
__global__ void __launch_bounds__(256) maxw_kernel(const float* __restrict__ W,
                                                   unsigned int* __restrict__ maxkey) {
    __shared__ unsigned int red[256];
    const size_t total4 = (size_t)N_NEURONS * N_NEURONS / 4;
    size_t i = (size_t)blockIdx.x * blockDim.x + threadIdx.x;
    const size_t stride = (size_t)gridDim.x * blockDim.x;
    float m = -3.4e38f;
    const float4* W4 = (const float4*)W;
    for (; i < total4; i += stride) {
        float4 v = W4[i];
        m = fmaxf(m, fmaxf(fmaxf(v.x, v.y), fmaxf(v.z, v.w)));
    }
    red[threadIdx.x] = float_key(m);
    __syncthreads();
    for (int w = 128; w > 0; w >>= 1) {
        if ((int)threadIdx.x < w)
            red[threadIdx.x] = max(red[threadIdx.x], red[threadIdx.x + w]);
        __syncthreads();
    }
    if (threadIdx.x == 0) atomicMax(maxkey, red[0]);
}

// ---------------------------------------------------------------------------
// Kernel 2: initial activations ACT[b][i] = sum_j W[i][j] * x[b][j]
// via V_WMMA_F32_16X16X4_F32, batch padded 8 -> 16.
// 4 independent accumulator chains so 8 loads can be in flight per wait
// (the 16x16x4 WMMA is short; this loop is L2-latency bound otherwise).
// ---------------------------------------------------------------------------
__global__ void __launch_bounds__(256) act_wmma_kernel(const float* __restrict__ x,
                                                       const float* __restrict__ W,
                                                       float* __restrict__ act) {
    const int lane = threadIdx.x & 31;
    const int wave = threadIdx.x >> 5;
    const int m    = lane & 15;   // D column = batch index (padded to 16)
    const int hi   = lane >> 4;   // lane half selects K/M split
    const int rowTile = blockIdx.x * 8 + wave;   // 8 waves per block
    const int row0 = rowTile * 16;

    const float* wrow  = W + (size_t)(row0 + m) * N_NEURONS + 2 * hi;
    const float  bmask = (m < BATCH) ? 1.0f : 0.0f;           // pad batch 8->16
    const float* xrow  = x + (size_t)(m & 7) * N_NEURONS + 2 * hi;

    v8f c0 = {}, c1 = {}, c2 = {}, c3 = {};
    for (int k = 0; k < N_NEURONS; k += 16) {
        v2f a0 = *(const v2f*)(wrow + k);
        v2f b0 = *(const v2f*)(xrow + k);
        v2f a1 = *(const v2f*)(wrow + k + 4);
        v2f b1 = *(const v2f*)(xrow + k + 4);
        v2f a2 = *(const v2f*)(wrow + k + 8);
        v2f b2 = *(const v2f*)(xrow + k + 8);
        v2f a3 = *(const v2f*)(wrow + k + 12);
        v2f b3 = *(const v2f*)(xrow + k + 12);
        b0 = b0 * bmask;
        b1 = b1 * bmask;
        b2 = b2 * bmask;
        b3 = b3 * bmask;
        c0 = __builtin_amdgcn_wmma_f32_16x16x4_f32(false, a0, false, b0, (short)0, c0, false, false);
        c1 = __builtin_amdgcn_wmma_f32_16x16x4_f32(false, a1, false, b1, (short)0, c1, false, false);
        c2 = __builtin_amdgcn_wmma_f32_16x16x4_f32(false, a2, false, b2, (short)0, c2, false, false);
        c3 = __builtin_amdgcn_wmma_f32_16x16x4_f32(false, a3, false, b3, (short)0, c3, false, false);
    }
    v8f c = (c0 + c1) + (c2 + c3);
    if (m < BATCH) {
        float* dst = act + (size_t)m * N_NEURONS + row0 + 8 * hi;
#pragma unroll
        for (int v = 0; v < 8; ++v) dst[v] = c[v];   // D[v+8*hi][m] -> row row0+8*hi+v
    }
}

// ---------------------------------------------------------------------------
// Kernel 3: sequential async sweeps, one 1024-thread workgroup per batch.
// State (s, a == W@s) is LDS-resident; initial staging via the Tensor Data
// Mover (TDM, one issuing wave, s_wait_tensorcnt + barrier). Shared writes
// only happen inside barrier-bracketed regions, and the take-the-branch
// decision is a deterministic function of barrier-consistent LDS state
// => block-uniform => barriers always match.
// ---------------------------------------------------------------------------
__global__ void __launch_bounds__(1024) sweep_kernel(const float* __restrict__ x,
                                                     const float* __restrict__ W,
                                                     const float* __restrict__ act,
                                                     const unsigned int* __restrict__ maxkey,
                                                     const int* __restrict__ niter_p,
                                                     float* __restrict__ out) {
    __shared__ float s_lds[N_NEURONS];   // 25.6 KB
    __shared__ float a_lds[N_NEURONS];   // 25.6 KB
    __shared__ float red[1024];          //  4.0 KB  (55.2 KB of 320 KB WGP LDS)

    const int tid = threadIdx.x;
    const int b   = blockIdx.x;

#if __has_builtin(__builtin_amdgcn_tensor_load_to_lds)
    if (tid == 0) {  // exactly one wave issues the two TDM descriptors
        tdm_load_1d((unsigned int)(uintptr_t)(void*)s_lds, x   + (size_t)b * N_NEURONS, N_NEURONS);
        tdm_load_1d((unsigned int)(uintptr_t)(void*)a_lds, act + (size_t)b * N_NEURONS, N_NEURONS);
        __builtin_amdgcn_s_wait_tensorcnt(0);
    }
#else
    for (int j = tid; j < N_NEURONS; j += 1024) {
        s_lds[j] = x[(size_t)b * N_NEURONS + j];
        a_lds[j] = act[(size_t)b * N_NEURONS + j];
    }
#endif
    __syncthreads();

    const float maxW  = key_float(*maxkey);
    const float scale = 4.0f / ((float)(N_NEURONS - 1) * maxW);
    const int num_iter = *niter_p;

    // initial energy: E = sum_j s_j * (1 - 0.5 * a_j)   (a == W s)
    float e_prev;
    {
        float p = 0.0f;
        for (int j = tid; j < N_NEURONS; j += 1024)
            p += s_lds[j] * (1.0f - 0.5f * a_lds[j]);
        red[tid] = p;
        __syncthreads();
        for (int w = 512; w > 0; w >>= 1) {
            if (tid < w) red[tid] += red[tid + w];
            __syncthreads();
        }
        e_prev = red[0];
        __syncthreads();   // everyone has read red[0] before any future reuse
    }

    bool done = false;
    for (int it = 0; it < num_iter && !done; ++it) {
        // hashed coprime-stride permutation of [0, 6400): step odd, step % 5 != 0
        unsigned int h1 = wang_hash((unsigned)(b * 977 + it * 131 + 1));
        unsigned int h2 = wang_hash(h1 ^ 0x9e3779b9u);
        unsigned int step = ((h1 >> 7) % (N_NEURONS / 2)) * 2u + 1u;
        if (step % 5u == 0u) step += 2u;        // stays odd, leaves residue-5 class
        unsigned int idx = h2 % N_NEURONS;

        for (int i = 0; i < N_NEURONS; ++i) {
            unsigned int nidx = idx + step;
            if (nidx >= N_NEURONS) nidx -= N_NEURONS;
            // prefetch next candidate W row (200 x 128B lines) L2 -> WGP$
            if (tid < 200)
                __builtin_prefetch(W + (size_t)nidx * N_NEURONS + tid * 32, 0, 0);

            const float aval = a_lds[idx];
            const float sold = s_lds[idx];
            // nearest of TIMESTEPS {-4..4}; argmin tie-break picks lower value
            float t = fminf(fmaxf(ceilf(aval * scale - 0.5f), -4.0f), 4.0f);
            const float delta = t - sold;   // identical across all threads

            if (delta != 0.0f) {            // block-uniform branch
                __syncthreads();            // all reads of old state complete
                if (tid == 0) s_lds[idx] = t;
                const float* wrow = W + (size_t)idx * N_NEURONS;
                for (int j = tid; j < N_NEURONS; j += 1024)
                    a_lds[j] = fmaf(wrow[j], delta, a_lds[j]);   // rank-1 update
                __syncthreads();            // updates visible before next step
            }
            idx = nidx;
        }

        // energy after sweep
        float p = 0.0f;
        for (int j = tid; j < N_NEURONS; j += 1024)
            p += s_lds[j] * (1.0f - 0.5f * a_lds[j]);
        red[tid] = p;
        __syncthreads();
        for (int w = 512; w > 0; w >>= 1) {
            if (tid < w) red[tid] += red[tid + w];
            __syncthreads();
        }
        const float e = red[0];
        __syncthreads();
        done = (e == e_prev);   // sticky freeze == early exit (state can't change)
        e_prev = e;
    }

    for (int j = tid; j < N_NEURONS; j += 1024)
        out[(size_t)b * N_NEURONS + j] = s_lds[j];
}

// ---------------------------------------------------------------------------
extern "C" void kernel_launch(void* const* d_in, const int* in_sizes, int n_in,
                              void* d_out, int out_size, void* d_ws, size_t ws_size,
                              hipStream_t stream) {
    const float* x     = (const float*)d_in[0];   // (8, 6400) f32
    const float* W     = (const float*)d_in[1];   // (6400, 6400) f32
    const int*   niter = (const int*)d_in[2];     // scalar
    float* out = (float*)d_out;                   // (8, 80, 80) f32

    unsigned int* maxkey = (unsigned int*)d_ws;
    float* act = (float*)((char*)d_ws + 256);     // 8 * 6400 f32 = 204.8 KB

    init_ws_kernel<<<1, 32, 0, stream>>>(maxkey);
    maxw_kernel<<<512, 256, 0, stream>>>(W, maxkey);
    // 400 row-tiles of 16, 8 waves (tiles) per 256-thread block -> 50 blocks
    act_wmma_kernel<<<(N_NEURONS / 16) / 8, 256, 0, stream>>>(x, W, act);
    sweep_kernel<<<BATCH, 1024, 0, stream>>>(x, W, act, maxkey, niter, out);
}